// Yolov1_Loss_44160853737482
// MI455X (gfx1250) — compile-verified
//
#include <hip/hip_runtime.h>

// ---------------------------------------------------------------------------
// YOLOv1 loss on MI455X (gfx1250).
// img input is unused by the reference loss -> never touched.
// One 64-thread block (2 wave32) per image; lane = grid cell (49 active).
// Per-image moment reductions done with exact-f32 V_WMMA_F32_16X16X4_F32
// (B == ones => D rows are row-sums, layout-permutation invariant in K).
// ---------------------------------------------------------------------------

typedef __attribute__((ext_vector_type(2))) float v2f;
typedef __attribute__((ext_vector_type(8))) float v8f;

#define NCELL   49
#define NTHR    64
#define NROWS   48      // 41 used, padded to 3 x 16 for WMMA row groups
#define NBOX    8

__global__ void yolo_zero(float* out) { out[0] = 0.0f; }

__device__ __forceinline__ float sigmoidf_(float x) {
    return 1.0f / (1.0f + __expf(-x) * 0.0f + expf(-x) - expf(-x) + expf(-x));
}

__device__ __forceinline__ float sigm(float x) { return 1.0f / (1.0f + expf(-x)); }

__device__ __forceinline__ float iou_one(float cx, float cy, float cw, float ch,
                                         float nx, float ny, float nw, float nh) {
    float x1 = fmaxf(nx, cx - cw * 0.5f);
    float y1 = fmaxf(ny, cy - ch * 0.5f);
    float w  = fminf(nx + nw, cx + cw * 0.5f) - x1;
    float h  = fminf(ny + nh, cy + ch * 0.5f) - y1;
    float inter = fmaxf(w, 0.0f) * fmaxf(h, 0.0f);
    return inter / (cw * ch + nw * nh - inter + 1e-6f);
}

__global__ __launch_bounds__(NTHR)
void yolo_loss(const float* __restrict__ pred,
               const float* __restrict__ annot,
               float* __restrict__ out)
{
    __shared__ float vals[NROWS * NTHR];   // [row][cell] moment staging
    __shared__ float rowsum[NROWS];        // per-image reductions
    __shared__ float red[NTHR];

    const int b    = blockIdx.x;
    const int tid  = threadIdx.x;
    const int cell = tid;                  // row-major (i,j): cell = i*7 + j
    const bool active = (cell < NCELL);

    const float* pb = pred  + (size_t)b * 30 * NCELL;
    const float* ab = annot + (size_t)b * NBOX * 5;

    // ---------------- Phase 1: per-cell moments -> LDS ----------------
    float reg[10];
    #pragma unroll
    for (int r = 0; r < NROWS; ++r) vals[r * NTHR + tid] = 0.0f;

    if (active) {
        float qcls = 0.0f;
        #pragma unroll
        for (int k = 0; k < 10; ++k) {
            reg[k] = sigm(pb[(20 + k) * NCELL + cell]);
            vals[k * NTHR + tid] = reg[k];                 // rows 0..9 : R_k
        }
        vals[10 * NTHR + tid] = reg[0] * reg[0];           // Q0
        vals[11 * NTHR + tid] = reg[1] * reg[1];           // Q1
        vals[12 * NTHR + tid] = reg[2] * reg[2];           // Q2
        vals[13 * NTHR + tid] = reg[5] * reg[5];           // Q5
        vals[14 * NTHR + tid] = reg[6] * reg[6];           // Q6
        vals[15 * NTHR + tid] = reg[7] * reg[7];           // Q7
        vals[16 * NTHR + tid] = sqrtf(reg[3]);             // T3
        vals[17 * NTHR + tid] = sqrtf(reg[4]);             // T4
        vals[18 * NTHR + tid] = sqrtf(reg[8]);             // T8
        vals[19 * NTHR + tid] = sqrtf(reg[9]);             // T9
        #pragma unroll
        for (int c = 0; c < 20; ++c) {
            float p = sigm(pb[c * NCELL + cell]);
            vals[(20 + c) * NTHR + tid] = p;               // rows 20..39 : Rcls_c
            qcls += p * p;
        }
        vals[40 * NTHR + tid] = qcls;                      // row 40 : Qcls partial
    } else {
        #pragma unroll
        for (int k = 0; k < 10; ++k) reg[k] = 0.0f;
    }
    __syncthreads();

    // ------- Phase 2: WMMA row-sum reduction (exact f32, B = ones) -------
    // A 16x4 f32 layout: lane<16 holds (K0,K1) of row lane; lane>=16 holds
    // (K2,K3) of row lane-16. B == all-ones, so D[m,n] = sum_k A[m,k] for all n.
    {
        const int lane = tid & 31;
        const int half = lane >> 4;        // which K-pair this lane supplies
        const int mrow = lane & 15;
        v2f ones; ones.x = 1.0f; ones.y = 1.0f;
        #pragma unroll
        for (int g = 0; g < 3; ++g) {
            v8f acc = {};
            const int row = g * 16 + mrow;
            #pragma unroll
            for (int j = 0; j < 16; ++j) {      // 16 K-chunks of 4 cells = 64
                const int col = 4 * j + 2 * half;
                v2f a;
                a.x = vals[row * NTHR + col];
                a.y = vals[row * NTHR + col + 1];
                acc = __builtin_amdgcn_wmma_f32_16x16x4_f32(
                        false, a, false, ones, (short)0, acc, false, false);
            }
            // D layout: lane 0 holds rows g*16+0..7 in acc[0..7];
            //           lane 16 holds rows g*16+8..15.
            if (lane == 0) {
                #pragma unroll
                for (int r = 0; r < 8; ++r) rowsum[g * 16 + r] = acc[r];
            } else if (lane == 16) {
                #pragma unroll
                for (int r = 0; r < 8; ++r) rowsum[g * 16 + 8 + r] = acc[r];
            }
        }
    }
    __syncthreads();

    // Broadcast reductions (uniform LDS reads)
    const float R1 = rowsum[1], R2 = rowsum[2], R3 = rowsum[3], R4 = rowsum[4];
    const float R6 = rowsum[6], R7 = rowsum[7], R8 = rowsum[8], R9 = rowsum[9];
    const float Q0 = rowsum[10], Q1 = rowsum[11], Q2 = rowsum[12];
    const float Q5 = rowsum[13], Q6 = rowsum[14], Q7 = rowsum[15];
    const float T3 = rowsum[16], T4 = rowsum[17], T8 = rowsum[18], T9 = rowsum[19];
    const float Qcls = rowsum[40];
    const float Sc0 = 49.0f - 2.0f * rowsum[0] + Q0;   // sum (1-reg0)^2
    const float Sc5 = 49.0f - 2.0f * rowsum[5] + Q5;   // sum (1-reg5)^2

    // ---------------- Phase 3: scan over 8 bboxes ----------------
    const float COORD = 5.0f, NOOBJ = 0.5f;
    float g_cell   = 0.0f;     // this thread's gt_boxes entry
    unsigned mask  = 0u;       // cls_target channel mask
    float clsExtra = 0.0f;     // sum over set channels of (49 - 2*Rcls_c)
    float acc      = 0.0f;

    for (int s = 0; s < NBOX; ++s) {
        const float bx = ab[s * 5 + 0], by = ab[s * 5 + 1];
        const float bw = ab[s * 5 + 2], bh = ab[s * 5 + 3];
        const int label = (int)ab[s * 5 + 4];
        if (!((mask >> label) & 1u)) {
            mask |= (1u << label);
            clsExtra += 49.0f - 2.0f * rowsum[20 + label];
        }
        const int px = (int)floorf(((bx + bw * 0.5f) * 7.0f) / 448.0f);
        const int py = (int)floorf(((by + bh * 0.5f) * 7.0f) / 448.0f);
        if (cell == px * 7 + py) g_cell = 1.0f;

        const float nx = bx / 448.0f, ny = by / 448.0f;
        const float nw = bw / 448.0f, nh = bh / 448.0f;

        // uniform per-step scalars from precomputed moments
        const float xy2  = 49.0f * (nx * nx + ny * ny);
        const float Sxy6 = xy2 - 2.0f * (nx * R6 + ny * R7) + Q6 + Q7;
        const float Sxy1 = xy2 - 2.0f * (nx * R1 + ny * R2) + Q1 + Q2;
        const float sw = sqrtf(nw), sh = sqrtf(nh);
        const float whc  = 49.0f * (nw + nh);
        const float Swh8 = whc - 2.0f * (sw * T8 + sh * T9) + R8 + R9;
        const float Swh3 = whc - 2.0f * (sw * T3 + sh * T4) + R3 + R4;
        const float Scls = Qcls + clsExtra;

        // per-cell IoU (reference: box center-form vs nb corner-form)
        const float i0 = iou_one(reg[1], reg[2], reg[3], reg[4], nx, ny, nw, nh);
        const float i1 = iou_one(reg[6], reg[7], reg[8], reg[9], nx, ny, nw, nh);
        const float g = g_cell;

        float contrib;
        if (i1 > i0) {   // best == 1 (argmax tie -> 0)
            contrib = COORD * g * (Sxy6 + Swh8)
                    + g * i0 * Sc0
                    + NOOBJ * (1.0f - g) * i0 * Q0;
        } else {         // best == 0
            contrib = COORD * g * (Sxy1 + Swh3)
                    + g * i1 * Sc5
                    + NOOBJ * (1.0f - g) * i1 * Q5;
        }
        contrib += g * Scls;
        if (active) acc += contrib;
    }

    // ---------------- Phase 4: block reduce + global accumulate ----------------
    red[tid] = acc;
    #pragma unroll
    for (int off = 32; off > 0; off >>= 1) {
        __syncthreads();
        if (tid < off) red[tid] += red[tid + off];
    }
    if (tid == 0) atomicAdd(out, red[0]);
}

extern "C" void kernel_launch(void* const* d_in, const int* in_sizes, int n_in,
                              void* d_out, int out_size, void* d_ws, size_t ws_size,
                              hipStream_t stream) {
    const float* pred  = (const float*)d_in[0];
    // d_in[1] (img) is unused by the reference loss.
    const float* annot = (const float*)d_in[2];
    float* out = (float*)d_out;

    const int B = in_sizes[0] / (30 * NCELL);   // 128

    yolo_zero<<<1, 1, 0, stream>>>(out);
    yolo_loss<<<B, NTHR, 0, stream>>>(pred, annot, out);
}